// GAT_29394756174152
// MI455X (gfx1250) — compile-verified
//
#include <hip/hip_runtime.h>
#include <math.h>

#define N_NODES 50000
#define N_EDGES 500000
#define HEADS 6
#define CH 64
#define FOUT 384   /* HEADS*CH */
#define IN_DIM 131
#define IN_PAD 132
#define MT_PER_WAVE 5   /* 3125 M-tiles = 5 * 625 */

typedef float v2f __attribute__((ext_vector_type(2)));
typedef float v8f __attribute__((ext_vector_type(8)));

// ---------------- padding kernels (layer 0 K: 131 -> 132) ----------------
__global__ void gat_pad_x(const float* __restrict__ x, float* __restrict__ xp) {
  long i = (long)blockIdx.x * blockDim.x + threadIdx.x;
  long total = (long)N_NODES * IN_PAD;
  if (i >= total) return;
  int n = (int)(i / IN_PAD), c = (int)(i % IN_PAD);
  xp[i] = (c < IN_DIM) ? x[(long)n * IN_DIM + c] : 0.0f;
}

__global__ void gat_pad_w0(const float* __restrict__ w, float* __restrict__ wp) {
  int i = blockIdx.x * blockDim.x + threadIdx.x;
  if (i >= IN_PAD * FOUT) return;
  int k = i / FOUT, c = i % FOUT;
  wp[i] = (k < IN_DIM) ? w[k * FOUT + c] : 0.0f;
}

// ---------------- fill ----------------
__global__ void gat_fill(float* __restrict__ p, int n, float v) {
  int i = blockIdx.x * blockDim.x + threadIdx.x;
  int stride = gridDim.x * blockDim.x;
  for (; i < n; i += stride) p[i] = v;
}

// ---------------- GEMM via V_WMMA_F32_16X16X4_F32 ----------------
// X[M x FOUT] = A[M x K] * B[K x FOUT].
// One wave computes MT_PER_WAVE (5) vertically-stacked 16x16 tiles sharing one
// B fragment per K-step: 5 A b64-loads + 2 B b32-loads per 5 WMMAs.
// Waves are nt-major so the 8 waves of a block share the same B tile (L0 hits).
// Grid: 24 n-tiles * 625 m-groups = 15000 waves (exact, EXEC always all-ones).
__global__ void gat_gemm_wmma(const float* __restrict__ A,
                              const float* __restrict__ B,
                              float* __restrict__ X, int K) {
  const int MG = (N_NODES / 16) / MT_PER_WAVE;  // 625
  int wv = (int)((blockIdx.x * blockDim.x + threadIdx.x) >> 5);
  int lane = threadIdx.x & 31;
  int nt = wv / MG;     // 0..23
  int mg = wv % MG;     // 0..624
  if (nt >= FOUT / 16) return;  // never taken with exact grid; wave-uniform
  int n0 = nt * 16;
  int half = lane >> 4;   // 0: K={0,1}, 1: K={2,3} within a K-step
  int l15 = lane & 15;

  v8f c[MT_PER_WAVE];
#pragma unroll
  for (int j = 0; j < MT_PER_WAVE; ++j) c[j] = (v8f){};

  // A fragment row for tile j: row = mg*80 + j*16 + l15, k-offset = half*2
  const float* arow[MT_PER_WAVE];
#pragma unroll
  for (int j = 0; j < MT_PER_WAVE; ++j)
    arow[j] = A + (long)(mg * (16 * MT_PER_WAVE) + j * 16 + l15) * K + half * 2;
  // B fragment: lane holds B[k+2*half][n0+l15], B[k+2*half+1][n0+l15]
  const float* bcol = B + (long)(half * 2) * FOUT + n0 + l15;

  for (int k = 0; k < K; k += 4) {
    v2f b;
    b.x = bcol[(long)k * FOUT];
    b.y = bcol[(long)(k + 1) * FOUT];
#pragma unroll
    for (int j = 0; j < MT_PER_WAVE; ++j) {
      const float2 av = *(const float2*)(arow[j] + k);
      v2f a; a.x = av.x; a.y = av.y;
      c[j] = __builtin_amdgcn_wmma_f32_16x16x4_f32(false, a, false, b,
                                                   (short)0, c[j], false, false);
    }
  }

  // C/D layout: VGPR i -> row base + i + 8*half, col n0 + l15
#pragma unroll
  for (int j = 0; j < MT_PER_WAVE; ++j) {
    float* xout = X + (long)(mg * (16 * MT_PER_WAVE) + j * 16 + half * 8) * FOUT
                    + n0 + l15;
#pragma unroll
    for (int i = 0; i < 8; ++i)
      xout[(long)i * FOUT] = c[j][i];
  }
}

// ---------------- attention logits: al_s/al_d[n,h] = sum_c x[n,h,c]*a[h,c] ----
__global__ void gat_attn_logits(const float* __restrict__ X,
                                const float* __restrict__ a_s,
                                const float* __restrict__ a_d,
                                float* __restrict__ als,
                                float* __restrict__ ald) {
  int w = (int)((blockIdx.x * blockDim.x + threadIdx.x) >> 5);  // node*HEADS+h
  int lane = threadIdx.x & 31;
  if (w >= N_NODES * HEADS) return;  // wave-uniform
  int n = w / HEADS, h = w % HEADS;
  const float* xv = X + (long)n * FOUT + h * CH;
  float x0 = xv[lane], x1 = xv[lane + 32];
  float s = x0 * a_s[h * CH + lane] + x1 * a_s[h * CH + lane + 32];
  float d = x0 * a_d[h * CH + lane] + x1 * a_d[h * CH + lane + 32];
#pragma unroll
  for (int off = 16; off > 0; off >>= 1) {
    s += __shfl_down(s, off, 32);
    d += __shfl_down(d, off, 32);
  }
  if (lane == 0) { als[w] = s; ald[w] = d; }
}

// ---------------- edge pass 1: leaky-relu logits + segment max ----------------
__global__ void gat_edge_lr_max(const int* __restrict__ src,
                                const int* __restrict__ dst,
                                const float* __restrict__ als,
                                const float* __restrict__ ald,
                                float* __restrict__ lrbuf,
                                float* __restrict__ m) {
  int i = blockIdx.x * blockDim.x + threadIdx.x;
  if (i >= N_EDGES * HEADS) return;
  int e = i / HEADS, h = i % HEADS;
  int s = src[e], d = dst[e];
  float v = als[s * HEADS + h] + ald[d * HEADS + h];
  float lr = (v > 0.0f) ? v : 0.2f * v;
  lrbuf[i] = lr;
  atomicMax(&m[d * HEADS + h], lr);  // global_atomic_max_num_f32
}

// ---------------- edge pass 2: exp + segment sum ----------------
__global__ void gat_edge_exp_sum(const int* __restrict__ dst,
                                 const float* __restrict__ m,
                                 float* __restrict__ exbuf,  // in: lr, out: exp
                                 float* __restrict__ denom) {
  int i = blockIdx.x * blockDim.x + threadIdx.x;
  if (i >= N_EDGES * HEADS) return;
  int e = i / HEADS, h = i % HEADS;
  int d = dst[e];
  float ex = __expf(exbuf[i] - m[d * HEADS + h]);
  exbuf[i] = ex;
  atomicAdd(&denom[d * HEADS + h], ex);
}

// ---------------- edge pass 3: weighted scatter-add (wave per edge*head) ------
__global__ void gat_edge_scatter(const int* __restrict__ src,
                                 const int* __restrict__ dst,
                                 const float* __restrict__ exbuf,
                                 const float* __restrict__ denom,
                                 const float* __restrict__ X,
                                 float* __restrict__ out) {
  int lane = threadIdx.x & 31;
  int wave = (int)((blockIdx.x * blockDim.x + threadIdx.x) >> 5);
  int nwaves = (int)((gridDim.x * blockDim.x) >> 5);
  for (int w = wave; w < N_EDGES * HEADS; w += nwaves) {
    int e = w / HEADS, h = w % HEADS;
    int s = src[e], d = dst[e];
    float alpha = exbuf[w] / (denom[d * HEADS + h] + 1e-16f);
    const float* xv = X + (long)s * FOUT + h * CH;
    float* ov = out + (long)d * FOUT + h * CH;
    atomicAdd(&ov[lane], alpha * xv[lane]);
    atomicAdd(&ov[lane + 32], alpha * xv[lane + 32]);
  }
}

// ---------------- epilogue: elu(out + b) in place (layers 0-2) ----------------
__global__ void gat_bias_elu(float* __restrict__ out, const float* __restrict__ b) {
  long i = (long)blockIdx.x * blockDim.x + threadIdx.x;
  if (i >= (long)N_NODES * FOUT) return;
  float v = out[i] + b[(int)(i % FOUT)];
  out[i] = (v > 0.0f) ? v : (__expf(v) - 1.0f);
}

// ---------------- final: mean over heads + b3 ----------------
__global__ void gat_final_mean(const float* __restrict__ out,
                               const float* __restrict__ b3,
                               float* __restrict__ y) {
  int i = blockIdx.x * blockDim.x + threadIdx.x;
  if (i >= N_NODES * CH) return;
  int n = i / CH, c = i % CH;
  float s = 0.0f;
#pragma unroll
  for (int h = 0; h < HEADS; ++h) s += out[(long)n * FOUT + h * CH + c];
  y[i] = s * (1.0f / HEADS) + b3[c];
}

static inline int ceil_div(long a, int b) { return (int)((a + b - 1) / b); }

extern "C" void kernel_launch(void* const* d_in, const int* in_sizes, int n_in,
                              void* d_out, int out_size, void* d_ws, size_t ws_size,
                              hipStream_t stream) {
  const float* x = (const float*)d_in[0];
  const int* edge_index = (const int*)d_in[1];
  const int* src = edge_index;
  const int* dst = edge_index + N_EDGES;
  const float* W[4]; const float* AS[4]; const float* AD[4]; const float* Bb[4];
  for (int l = 0; l < 4; ++l) {
    W[l]  = (const float*)d_in[2 + 4 * l];
    AS[l] = (const float*)d_in[3 + 4 * l];
    AD[l] = (const float*)d_in[4 + 4 * l];
    Bb[l] = (const float*)d_in[5 + 4 * l];
  }
  float* y = (float*)d_out;

  // workspace carve-up (floats)
  float* ws = (float*)d_ws;
  float* xpad  = ws;                 ws += (long)N_NODES * IN_PAD;
  float* w0pad = ws;                 ws += (long)IN_PAD * FOUT;
  float* bufX  = ws;                 ws += (long)N_NODES * FOUT;     // x = h@W
  float* bufH  = ws;                 ws += (long)N_NODES * FOUT;
  float* bufO  = ws;                 ws += (long)N_NODES * FOUT;
  float* als   = ws;                 ws += (long)N_NODES * HEADS;
  float* ald   = ws;                 ws += (long)N_NODES * HEADS;
  float* segm  = ws;                 ws += (long)N_NODES * HEADS;
  float* denom = ws;                 ws += (long)N_NODES * HEADS;
  float* exbuf = ws;                 ws += (long)N_EDGES * HEADS;

  const int T = 256;

  // pad layer-0 operands
  gat_pad_x<<<ceil_div((long)N_NODES * IN_PAD, T), T, 0, stream>>>(x, xpad);
  gat_pad_w0<<<ceil_div((long)IN_PAD * FOUT, T), T, 0, stream>>>(W[0], w0pad);

  const int gemm_waves = (FOUT / 16) * ((N_NODES / 16) / MT_PER_WAVE);  // 15000
  const int gemm_blocks = gemm_waves / (T / 32);             // 1875 (exact)
  const int attn_blocks = (N_NODES * HEADS) / (T / 32);      // 37500 (exact)
  const int edge_blocks = ceil_div((long)N_EDGES * HEADS, T);
  const int nh = N_NODES * HEADS;
  const int nf = N_NODES * FOUT;

  for (int l = 0; l < 4; ++l) {
    const float* hin = (l == 0) ? xpad : ((l == 1 || l == 3) ? bufO : bufH);
    float* outb = (l == 0 || l == 2) ? bufO : bufH;
    const float* Wl = (l == 0) ? w0pad : W[l];
    int K = (l == 0) ? IN_PAD : FOUT;

    // dense projection (WMMA) + per-node attention logits
    gat_gemm_wmma<<<gemm_blocks, T, 0, stream>>>(hin, Wl, bufX, K);
    gat_attn_logits<<<attn_blocks, T, 0, stream>>>(bufX, AS[l], AD[l], als, ald);

    // init segment buffers + output accumulator
    gat_fill<<<1024, T, 0, stream>>>(segm, nh, -INFINITY);
    gat_fill<<<1024, T, 0, stream>>>(denom, nh, 0.0f);
    gat_fill<<<4096, T, 0, stream>>>(outb, nf, 0.0f);

    // edge-wise softmax + aggregation
    gat_edge_lr_max<<<edge_blocks, T, 0, stream>>>(src, dst, als, ald, exbuf, segm);
    gat_edge_exp_sum<<<edge_blocks, T, 0, stream>>>(dst, segm, exbuf, denom);
    gat_edge_scatter<<<9600, T, 0, stream>>>(src, dst, exbuf, denom, bufX, outb);

    if (l < 3) {
      gat_bias_elu<<<ceil_div((long)nf, T), T, 0, stream>>>(outb, Bb[l]);
    } else {
      gat_final_mean<<<ceil_div((long)N_NODES * CH, T), T, 0, stream>>>(outb, Bb[3], y);
    }
  }
}